// ZBLRepulsion_48507360641462
// MI455X (gfx1250) — compile-verified
//
#include <hip/hip_runtime.h>
#include <hip/hip_bf16.h>
#include <math.h>

// ---------------------------------------------------------------------------
// ZBL repulsion energy, MI455X (gfx1250, wave32).
//  - Stage 1: pack per-atom {Z, Z^0.23} into L2-resident float2 table (ws).
//  - Stage 2: grid-stride over edges, 4 edges/iteration via b128 nontemporal
//             streams (20 B/edge HBM-bound), L2-resident zpack gathers,
//             per-wave exact f32 reduction via V_WMMA_F32_16X16X4_F32.
//  - Stage 3: single-block deterministic fold of block partials + final scale.
// ---------------------------------------------------------------------------

typedef __attribute__((ext_vector_type(2))) float v2f;
typedef __attribute__((ext_vector_type(4))) float v4f;
typedef __attribute__((ext_vector_type(8))) float v8f;
typedef __attribute__((ext_vector_type(4))) int   v4i;

// Exact f32 sum of all 32 lanes, broadcast to every lane.
// Uses V_WMMA_F32_16X16X4_F32 (K=4, f32 in/out, RNE) twice:
//   1) A.vgpr0 = x, A.vgpr1 = 0, B = ones  -> D[m][n] = x[m] + x[m+16]
//      per-lane sum of D's 8 VGPRs leaves   lanes 0..15 : sum(x[0..7],x[16..23])
//                                           lanes 16..31: sum(x[8..15],x[24..31])
//   2) same trick folds the two half-wave sums -> total in every lane of D2[0].
// EXEC must be all ones at the call site (callers guarantee convergence).
__device__ __forceinline__ float wave32_sum_f32_wmma(float x) {
  v2f a;    a.x = x;    a.y = 0.0f;
  v2f ones; ones.x = 1.0f; ones.y = 1.0f;
  v8f c = {};
  v8f d = __builtin_amdgcn_wmma_f32_16x16x4_f32(
      /*neg_a=*/false, a, /*neg_b=*/false, ones,
      /*c_mod=*/(short)0, c, /*reuse_a=*/false, /*reuse_b=*/false);
  float t = ((d[0] + d[1]) + (d[2] + d[3])) + ((d[4] + d[5]) + (d[6] + d[7]));
  v2f a2; a2.x = t; a2.y = 0.0f;
  v8f d2 = __builtin_amdgcn_wmma_f32_16x16x4_f32(
      false, a2, false, ones, (short)0, c, false, false);
  return d2[0];
}

__device__ __forceinline__ float softplus_precise(float x) {
  return logf(1.0f + expf(x));
}

// One edge's energy contribution (masked for self-edges).
__device__ __forceinline__ float edge_term(float dx, float dy, float dz,
                                           int ii, int jj,
                                           const float2* __restrict__ zpack,
                                           float c0, float c1, float c2, float c3) {
  const float INV_ANUM   = 1.0f / 0.4685f;          // 1/A_NUM
  const float HALF_PI_RM = 1.57079632679489662f;    // pi/R_MAX (R_MAX=2)
  float dr = sqrtf(dx * dx + dy * dy + dz * dz);
  dr = fminf(fmaxf(dr, 0.02f), 2.0f);
  float cosc = 0.5f * (__cosf(HALF_PI_RM * dr) + 1.0f);
  float2 zi = zpack[ii];
  float2 zj = zpack[jj];
  float dist = dr * (zi.y + zj.y) * INV_ANUM;
  float f = c0 * __expf(-3.19980f * dist)
          + c1 * __expf(-0.94229f * dist)
          + c2 * __expf(-0.40290f * dist)
          + c3 * __expf(-0.20162f * dist);
  float Eij = zi.x * zj.x * __builtin_amdgcn_rcpf(dr) * f * cosc;
  return (ii != jj) ? Eij : 0.0f;
}

// ---- Stage 1: per-atom pack {Z, Z^0.23} --------------------------------------
__global__ void zbl_prep_kernel(const float* __restrict__ Z,
                                float2* __restrict__ zpack, int natoms) {
  int i = blockIdx.x * blockDim.x + threadIdx.x;
  if (i < natoms) {
    float z = Z[i];
    zpack[i] = make_float2(z, __powf(z, 0.23f));
  }
}

// ---- Stage 2 (fast path, nedges % 4 == 0): 4 edges / iteration ---------------
// All streams are 16B-aligned b128 nontemporal loads; indices are 32-bit.
__global__ void __launch_bounds__(256)
zbl_main_vec4_kernel(const v4f* __restrict__ drv4,    // dr_vec as float4[3*nq]
                     const v4i* __restrict__ idx_i4,  // idx[0] as int4[nq]
                     const v4i* __restrict__ idx_j4,  // idx[1] as int4[nq]
                     const float2* __restrict__ zpack,
                     const float* __restrict__ coeffs_raw,
                     float* __restrict__ partials,
                     int nquads) {
  const float c0 = softplus_precise(coeffs_raw[0]);
  const float c1 = softplus_precise(coeffs_raw[1]);
  const float c2 = softplus_precise(coeffs_raw[2]);
  const float c3 = softplus_precise(coeffs_raw[3]);

  float acc = 0.0f;
  const int stride = gridDim.x * blockDim.x;
  for (int q = blockIdx.x * blockDim.x + threadIdx.x; q < nquads; q += stride) {
    v4i vi = __builtin_nontemporal_load(&idx_i4[q]);
    v4i vj = __builtin_nontemporal_load(&idx_j4[q]);
    v4f d0 = __builtin_nontemporal_load(&drv4[3 * q + 0]);
    v4f d1 = __builtin_nontemporal_load(&drv4[3 * q + 1]);
    v4f d2 = __builtin_nontemporal_load(&drv4[3 * q + 2]);
    // edge layout in 12 packed floats: e0=(d0.x,d0.y,d0.z) e1=(d0.w,d1.x,d1.y)
    //                                  e2=(d1.z,d1.w,d2.x) e3=(d2.y,d2.z,d2.w)
    acc += edge_term(d0.x, d0.y, d0.z, vi.x, vj.x, zpack, c0, c1, c2, c3);
    acc += edge_term(d0.w, d1.x, d1.y, vi.y, vj.y, zpack, c0, c1, c2, c3);
    acc += edge_term(d1.z, d1.w, d2.x, vi.z, vj.z, zpack, c0, c1, c2, c3);
    acc += edge_term(d2.y, d2.z, d2.w, vi.w, vj.w, zpack, c0, c1, c2, c3);
  }

  // All threads reconverged; full 256-thread blocks -> EXEC all ones for WMMA.
  float wsum = wave32_sum_f32_wmma(acc);

  __shared__ float smem[8];
  int wid = threadIdx.x >> 5;
  if ((threadIdx.x & 31) == 0) smem[wid] = wsum;
  __syncthreads();
  if (threadIdx.x == 0) {
    float s = 0.0f;
    int nw = blockDim.x >> 5;
    for (int w = 0; w < nw; ++w) s += smem[w];
    partials[blockIdx.x] = s;
  }
}

// ---- Stage 2 (general fallback): scalar per-edge -----------------------------
__global__ void __launch_bounds__(256)
zbl_main_scalar_kernel(const float* __restrict__ dr_vec,
                       const int* __restrict__ idx,
                       const float2* __restrict__ zpack,
                       const float* __restrict__ coeffs_raw,
                       float* __restrict__ partials,
                       long long nedges) {
  const float c0 = softplus_precise(coeffs_raw[0]);
  const float c1 = softplus_precise(coeffs_raw[1]);
  const float c2 = softplus_precise(coeffs_raw[2]);
  const float c3 = softplus_precise(coeffs_raw[3]);

  float acc = 0.0f;
  const long long stride = (long long)gridDim.x * blockDim.x;
  for (long long e = (long long)blockIdx.x * blockDim.x + threadIdx.x;
       e < nedges; e += stride) {
    int ii = idx[e];
    int jj = idx[nedges + e];
    acc += edge_term(dr_vec[3 * e + 0], dr_vec[3 * e + 1], dr_vec[3 * e + 2],
                     ii, jj, zpack, c0, c1, c2, c3);
  }

  float wsum = wave32_sum_f32_wmma(acc);

  __shared__ float smem[8];
  int wid = threadIdx.x >> 5;
  if ((threadIdx.x & 31) == 0) smem[wid] = wsum;
  __syncthreads();
  if (threadIdx.x == 0) {
    float s = 0.0f;
    int nw = blockDim.x >> 5;
    for (int w = 0; w < nw; ++w) s += smem[w];
    partials[blockIdx.x] = s;
  }
}

// ---- Stage 3: fold partials, apply 0.5*softplus(rep_scale) -------------------
__global__ void __launch_bounds__(256)
zbl_finish_kernel(const float* __restrict__ partials, int nparts,
                  const float* __restrict__ rep_scale,
                  float* __restrict__ out) {
  float acc = 0.0f;
  for (int i = threadIdx.x; i < nparts; i += blockDim.x) acc += partials[i];

  float wsum = wave32_sum_f32_wmma(acc);

  __shared__ float smem[8];
  int wid = threadIdx.x >> 5;
  if ((threadIdx.x & 31) == 0) smem[wid] = wsum;
  __syncthreads();
  if (threadIdx.x == 0) {
    float total = 0.0f;
    int nw = blockDim.x >> 5;
    for (int w = 0; w < nw; ++w) total += smem[w];
    float rs = softplus_precise(rep_scale[0]);
    out[0] = 0.5f * rs * total;
  }
}

// ---------------------------------------------------------------------------
extern "C" void kernel_launch(void* const* d_in, const int* in_sizes, int n_in,
                              void* d_out, int out_size, void* d_ws, size_t ws_size,
                              hipStream_t stream) {
  const float* dr_vec       = (const float*)d_in[0];   // [E*3] f32
  const float* Z            = (const float*)d_in[1];   // [natoms] f32
  const int*   idx          = (const int*)d_in[2];     // [2*E] i32
  const float* coefficients = (const float*)d_in[3];   // [4] f32
  const float* rep_scale    = (const float*)d_in[4];   // [1] f32

  const int natoms = in_sizes[1];
  const long long nedges = (long long)in_sizes[2] / 2;

  // workspace layout: [ zpack: natoms*float2 | partials: NB*float ]
  float2* zpack = (float2*)d_ws;
  size_t zbytes = ((size_t)natoms * sizeof(float2) + 255) & ~(size_t)255;
  float* partials = (float*)((char*)d_ws + zbytes);

  const int NT = 256;   // 8 wave32 per block
  const int NB = 2048;  // fixed grid -> deterministic partial sums

  zbl_prep_kernel<<<(natoms + NT - 1) / NT, NT, 0, stream>>>(Z, zpack, natoms);

  if ((nedges & 3LL) == 0 && nedges <= 0x7FFFFFFFLL) {
    // Fast path: all three streams are 16B-aligned -> pure b128 NT loads.
    const int nquads = (int)(nedges >> 2);
    zbl_main_vec4_kernel<<<NB, NT, 0, stream>>>(
        (const v4f*)dr_vec, (const v4i*)idx, (const v4i*)(idx + nedges),
        zpack, coefficients, partials, nquads);
  } else {
    zbl_main_scalar_kernel<<<NB, NT, 0, stream>>>(
        dr_vec, idx, zpack, coefficients, partials, nedges);
  }

  zbl_finish_kernel<<<1, NT, 0, stream>>>(partials, NB, rep_scale,
                                          (float*)d_out);
}